// GATLayer_1082331758609
// MI455X (gfx1250) — compile-verified
//
#include <hip/hip_runtime.h>

#define N_NODES   100000
#define N_EDGES   1600000
#define IN_FEATS  128
#define NUM_HEADS 4
#define OUT_FEATS 16
#define HF        64   // NUM_HEADS * OUT_FEATS
#define NEG_SLOPE 0.2f

typedef __attribute__((ext_vector_type(2))) float v2f;
typedef __attribute__((ext_vector_type(8))) float v8f;

// Order-preserving float<->uint mapping so atomicMax(u32) == IEEE float max.
__device__ __forceinline__ unsigned f2ord(float f) {
    unsigned u = __float_as_uint(f);
    return (u & 0x80000000u) ? ~u : (u | 0x80000000u);
}
__device__ __forceinline__ float ord2f(unsigned o) {
    unsigned u = (o & 0x80000000u) ? (o & 0x7FFFFFFFu) : ~o;
    return __uint_as_float(u);
}

// One block = one 16-row tile of nodes. 8 waves:
//   waves 0..3: h-tile for head w  (h = feats@W), plus el/er reductions
//   waves 4..7: residual tile for head w-4 (feats@res_W + bias) -> d_out base
__global__ __launch_bounds__(256) void gat_proj_kernel(
    const float* __restrict__ feats, const float* __restrict__ W,
    const float* __restrict__ resW,  const float* __restrict__ attn_l,
    const float* __restrict__ attn_r,const float* __restrict__ bias,
    float* __restrict__ h_ws, float* __restrict__ el, float* __restrict__ er,
    float* __restrict__ out)
{
    const int lane  = threadIdx.x & 31;
    const int wave  = threadIdx.x >> 5;
    const int mbase = blockIdx.x * 16;
    const bool is_res = (wave >= 4);
    const int hd   = is_res ? (wave - 4) : wave;
    const float* Bm = is_res ? resW : W;

    const int col  = lane & 15;            // N within 16-wide tile (= feat in head)
    const int koff = (lane < 16) ? 0 : 2;  // K sub-offset per ISA A/B layout

    int arow = mbase + col;                // A rows indexed by lane&15
    if (arow >= N_NODES) arow = 0;         // clamp; stores are guarded
    const float* Aptr = feats + (size_t)arow * IN_FEATS + koff;
    const float* Bptr = Bm + (size_t)koff * HF + hd * OUT_FEATS + col;

    v8f c = {};
    #pragma unroll
    for (int kk = 0; kk < IN_FEATS / 4; ++kk) {
        v2f a, b;
        a.x = Aptr[kk * 4 + 0];            // (m, 4kk+koff)
        a.y = Aptr[kk * 4 + 1];            // (m, 4kk+koff+1)
        b.x = Bptr[(size_t)(kk * 4 + 0) * HF];  // (4kk+koff,   n)
        b.y = Bptr[(size_t)(kk * 4 + 1) * HF];  // (4kk+koff+1, n)
        c = __builtin_amdgcn_wmma_f32_16x16x4_f32(
                /*neg_a=*/false, a, /*neg_b=*/false, b,
                /*c_mod=*/(short)0, c, /*reuse_a=*/false, /*reuse_b=*/false);
    }

    if (!is_res) {
        const float al = attn_l[hd * OUT_FEATS + col];
        const float ar = attn_r[hd * OUT_FEATS + col];
        #pragma unroll
        for (int r = 0; r < 8; ++r) {
            const int row = mbase + r + ((lane < 16) ? 0 : 8);
            const float cv = c[r];
            if (row < N_NODES)
                h_ws[(size_t)row * HF + hd * OUT_FEATS + col] = cv;
            // reduce over the 16 columns (stays within each 16-lane half)
            float sl = cv * al, sr = cv * ar;
            #pragma unroll
            for (int off = 8; off >= 1; off >>= 1) {
                sl += __shfl_xor(sl, off, 32);
                sr += __shfl_xor(sr, off, 32);
            }
            if (col == 0 && row < N_NODES) {
                el[(size_t)row * NUM_HEADS + hd] = sl;
                er[(size_t)row * NUM_HEADS + hd] = sr;
            }
        }
    } else {
        const float bv = bias[hd * OUT_FEATS + col];
        #pragma unroll
        for (int r = 0; r < 8; ++r) {
            const int row = mbase + r + ((lane < 16) ? 0 : 8);
            if (row < N_NODES)
                out[(size_t)row * HF + hd * OUT_FEATS + col] = c[r] + bv;
        }
    }
}

__global__ void gat_init_kernel(unsigned* __restrict__ m, float* __restrict__ denom) {
    const int i = blockIdx.x * blockDim.x + threadIdx.x;
    if (i < N_NODES * NUM_HEADS) { m[i] = 0u; denom[i] = 0.0f; }
}

// Pass 1: e = leaky_relu(el[src]+er[dst]); segment max into m via ordered-uint atomicMax.
__global__ void gat_edge_max_kernel(
    const int* __restrict__ src, const int* __restrict__ dst,
    const float* __restrict__ el, const float* __restrict__ er,
    float* __restrict__ e_ws, unsigned* __restrict__ m)
{
    const int i = blockIdx.x * blockDim.x + threadIdx.x;
    if (i >= N_EDGES * NUM_HEADS) return;
    const int e = i >> 2, hd = i & 3;
    const int s = src[e], d = dst[e];
    float v = el[(size_t)s * NUM_HEADS + hd] + er[(size_t)d * NUM_HEADS + hd];
    v = (v > 0.0f) ? v : NEG_SLOPE * v;
    e_ws[i] = v;
    atomicMax(m + (size_t)d * NUM_HEADS + hd, f2ord(v));
}

// Pass 2: e_exp = exp(e - m[dst]); segment sum into denom.
__global__ void gat_edge_exp_kernel(
    const int* __restrict__ dst, const unsigned* __restrict__ m,
    float* __restrict__ e_ws, float* __restrict__ denom)
{
    const int i = blockIdx.x * blockDim.x + threadIdx.x;
    if (i >= N_EDGES * NUM_HEADS) return;
    const int e = i >> 2, hd = i & 3;
    const int d = dst[e];
    const float mx = ord2f(m[(size_t)d * NUM_HEADS + hd]);
    const float ex = expf(e_ws[i] - mx);
    e_ws[i] = ex;
    atomicAdd(denom + (size_t)d * NUM_HEADS + hd, ex);
}

// Pass 3: alpha = e_exp/denom[dst]; out[dst] += alpha * h[src]  (16 floats per (e,hd)).
__global__ void gat_scatter_kernel(
    const int* __restrict__ src, const int* __restrict__ dst,
    const float* __restrict__ h_ws, const float* __restrict__ e_ws,
    const float* __restrict__ denom, float* __restrict__ out)
{
    const int i = blockIdx.x * blockDim.x + threadIdx.x;
    if (i >= N_EDGES * NUM_HEADS) return;
    const int e = i >> 2, hd = i & 3;
    const int s = src[e], d = dst[e];
    const float alpha = e_ws[i] / denom[(size_t)d * NUM_HEADS + hd];
    const float4* hp = (const float4*)(h_ws + (size_t)s * HF + hd * OUT_FEATS);
    float* op = out + (size_t)d * HF + hd * OUT_FEATS;
    #pragma unroll
    for (int q = 0; q < 4; ++q) {
        const float4 hv = hp[q];
        atomicAdd(op + q * 4 + 0, hv.x * alpha);
        atomicAdd(op + q * 4 + 1, hv.y * alpha);
        atomicAdd(op + q * 4 + 2, hv.z * alpha);
        atomicAdd(op + q * 4 + 3, hv.w * alpha);
    }
}

extern "C" void kernel_launch(void* const* d_in, const int* in_sizes, int n_in,
                              void* d_out, int out_size, void* d_ws, size_t ws_size,
                              hipStream_t stream) {
    (void)in_sizes; (void)n_in; (void)out_size; (void)ws_size;
    const float* feats  = (const float*)d_in[0];
    const int*   src    = (const int*)  d_in[1];
    const int*   dst    = (const int*)  d_in[2];
    const float* W      = (const float*)d_in[3];
    const float* attn_l = (const float*)d_in[4];
    const float* attn_r = (const float*)d_in[5];
    const float* resW   = (const float*)d_in[6];
    const float* bias   = (const float*)d_in[7];
    float* out = (float*)d_out;

    // Workspace layout (floats): h[N*64] | el[N*4] | er[N*4] | m[N*4] | denom[N*4] | e_ws[E*4]
    float* ws    = (float*)d_ws;
    float* h_ws  = ws;
    float* el    = h_ws + (size_t)N_NODES * HF;
    float* er    = el   + (size_t)N_NODES * NUM_HEADS;
    unsigned* m  = (unsigned*)(er + (size_t)N_NODES * NUM_HEADS);
    float* denom = (float*)(m + (size_t)N_NODES * NUM_HEADS);
    float* e_ws  = denom + (size_t)N_NODES * NUM_HEADS;

    const int mtiles = (N_NODES + 15) / 16;
    gat_proj_kernel<<<mtiles, 256, 0, stream>>>(feats, W, resW, attn_l, attn_r, bias,
                                                h_ws, el, er, out);

    const int nh = N_NODES * NUM_HEADS;
    gat_init_kernel<<<(nh + 255) / 256, 256, 0, stream>>>(m, denom);

    const int eh = N_EDGES * NUM_HEADS;
    gat_edge_max_kernel<<<(eh + 255) / 256, 256, 0, stream>>>(src, dst, el, er, e_ws, m);
    gat_edge_exp_kernel<<<(eh + 255) / 256, 256, 0, stream>>>(dst, m, e_ws, denom);
    gat_scatter_kernel<<<(eh + 255) / 256, 256, 0, stream>>>(src, dst, h_ws, e_ws, denom, out);
}